// LinearStringEncoder_39152921870348
// MI455X (gfx1250) — compile-verified
//
#include <hip/hip_runtime.h>

typedef __attribute__((ext_vector_type(2))) float v2f;
typedef __attribute__((ext_vector_type(8))) float v8f;

#define VOCAB  100000
#define HIDDEN 128
#define BATCH  1024
#define SEQLEN 200

// ---------------------------------------------------------------------------
// One-time transpose: W[h, v] (row-major, [128 x 100000]) -> Wt[v, h]
// so that each gathered vocab column becomes 512 contiguous bytes.
// Tile: 128 h x 32 v per block, staged through padded LDS (stride 33 -> no
// bank conflicts on the strided read-out).
// ---------------------------------------------------------------------------
__global__ __launch_bounds__(256) void transpose_w_kernel(
    const float* __restrict__ W, float* __restrict__ Wt) {
  __shared__ float tile[128 * 33];
  const int v0 = blockIdx.x * 32;
  const int t  = threadIdx.x;

#pragma unroll
  for (int i = 0; i < 16; ++i) {          // 256 thr * 16 = 4096 = 128*32
    int idx = t + i * 256;
    int h   = idx >> 5;                   // 0..127
    int vl  = idx & 31;                   // contiguous per 32 lanes
    tile[h * 33 + vl] = W[h * VOCAB + v0 + vl];
  }
  __syncthreads();
#pragma unroll
  for (int i = 0; i < 16; ++i) {
    int idx = t + i * 256;
    int vl  = idx >> 7;                   // 0..31
    int h   = idx & 127;                  // contiguous store of 128 floats
    Wt[(v0 + vl) * HIDDEN + h] = tile[h * 33 + vl];
  }
}

// ---------------------------------------------------------------------------
// Gather-sum encoder. One block (256 thr = 8 waves) per batch row.
// Wave w owns hidden chunk [16w, 16w+16). Per iteration it loads a 4x16 f32
// block B (4 gathered word-rows x 16 hidden) and accumulates with
// V_WMMA_F32_16X16X4_F32 using an all-ones A (exact in f32):
//     C[m, n] += sum_k B[k, n]   for every m.
// Slot map: lane column n = hidden%16; half-wave 'half' carries words
// (l+2*half, l+2*half+1) in (B.x, B.y). Per column n the four (vgpr, half)
// slots cover words l..l+3 exactly once, and with A==1 any bijective
// slot->K hardware mapping yields the correct column sum.
// After 50 WMMAs, acc VGPR0 lanes 0..15 hold y[b, 16w..16w+15].
//
// All gather indices are 32-bit element offsets (max id*128+h = 12.8M),
// so loads lower to global_load_b32 v, voff, s[base] scale_offset with no
// 64-bit VALU address math in the loop.
// ---------------------------------------------------------------------------
template <bool TRANSPOSED>
__global__ __launch_bounds__(256) void encoder_gather_wmma_kernel(
    const int* __restrict__ ids, const float* __restrict__ Wsrc,
    const float* __restrict__ bias, float* __restrict__ out) {
  __shared__ alignas(8) int sIds[SEQLEN];

  const int b = blockIdx.x;
  const int t = threadIdx.x;
  if (t < SEQLEN) sIds[t] = ids[b * SEQLEN + t];
  __syncthreads();

  const int lane = t & 31;
  const int wave = t >> 5;
  const int hc   = wave * 16;       // hidden chunk base for this wave
  const int n    = lane & 15;       // tile column = hidden offset
  const int half = lane >> 4;       // 0 or 1
  const int h    = hc + n;

  v8f acc = {};
  const v2f ones = {1.0f, 1.0f};    // A-matrix: 16x4 of exact 1.0f

  for (int l = 0; l < SEQLEN; l += 4) {
    // one ds_load_b64: two adjacent word ids for this half-wave
    const int2 p = *(const int2*)(&sIds[l + 2 * half]);
    v2f bmat;
    if (TRANSPOSED) {
      bmat.x = Wsrc[p.x * HIDDEN + h];     // 64B-contiguous per half-wave
      bmat.y = Wsrc[p.y * HIDDEN + h];
    } else {
      bmat.x = Wsrc[h * VOCAB + p.x];      // fallback, strided
      bmat.y = Wsrc[h * VOCAB + p.y];
    }
    acc = __builtin_amdgcn_wmma_f32_16x16x4_f32(
        /*neg_a=*/false, ones, /*neg_b=*/false, bmat,
        /*c_mod=*/(short)0, acc, /*reuse_a=*/false, /*reuse_b=*/false);
  }

  if (lane < 16) {
    out[b * HIDDEN + h] = acc[0] + bias[h];      // C row M=0, N=lane
  }
}

// ---------------------------------------------------------------------------
extern "C" void kernel_launch(void* const* d_in, const int* in_sizes, int n_in,
                              void* d_out, int out_size, void* d_ws,
                              size_t ws_size, hipStream_t stream) {
  (void)in_sizes; (void)n_in; (void)out_size;
  const int*   ids  = (const int*)d_in[0];    // [1024, 200] int32
  const float* W    = (const float*)d_in[1];  // [128, 100000] f32
  const float* bias = (const float*)d_in[2];  // [128] f32
  float*       out  = (float*)d_out;          // [1024, 128] f32

  const size_t need = (size_t)VOCAB * HIDDEN * sizeof(float);  // 51.2 MB
  if (ws_size >= need) {
    float* Wt = (float*)d_ws;
    transpose_w_kernel<<<VOCAB / 32, 256, 0, stream>>>(W, Wt);
    encoder_gather_wmma_kernel<true><<<BATCH, 256, 0, stream>>>(ids, Wt, bias, out);
  } else {
    encoder_gather_wmma_kernel<false><<<BATCH, 256, 0, stream>>>(ids, W, bias, out);
  }
}